// DifferentialMaxtree_12008728559978
// MI455X (gfx1250) — compile-verified
//
#include <hip/hip_runtime.h>
#include <math.h>

typedef __attribute__((ext_vector_type(2))) float v2f;
typedef __attribute__((ext_vector_type(8))) float v8f;

#define FEPS 1e-10f

// ---------------------------------------------------------------------------
// Kernel 1: per-node feature rescale -> 17-dim dot(weight) via WMMA f32
// 16x16x4 -> sigmoid -> term = diff * score.  Also materializes par[] with
// the -1 -> N sentinel remap, and writes the N-th sentinel slot.
// ---------------------------------------------------------------------------
__global__ __launch_bounds__(256) void k_term_par(
    const float* __restrict__ diff,
    const float* __restrict__ attr,
    const float* __restrict__ weight,
    const float* __restrict__ bias,
    const int*   __restrict__ parent,
    float* __restrict__ term0,
    int*   __restrict__ par0,
    int N)
{
    __shared__ float Xs[256][20];   // rescaled features, padded K=20
    __shared__ float Ds[256];       // maxtree_diff per node
    __shared__ float Ws[20];        // weight padded to 20 (zeros past 17)

    const int tid   = threadIdx.x;
    const int node  = blockIdx.x * 256 + tid;
    const int nodeC = (node < N) ? node : (N - 1);   // clamp so EXEC stays full

    if (blockIdx.x == 0 && tid == 0) { term0[N] = 0.0f; par0[N] = N; }
    if (tid < 20) Ws[tid] = (tid < 17) ? weight[tid] : 0.0f;

    // ---- phase 1: one node per lane, rescale features into LDS ----
    float f[15];
    #pragma unroll
    for (int c = 0; c < 15; ++c) f[c] = attr[(size_t)nodeC * 15 + c];

    float x[20];
    x[0] = f[0]; x[1] = f[1]; x[2] = f[2]; x[3] = f[3]; x[4] = f[4];
    #pragma unroll
    for (int c = 0; c < 9; ++c) x[5 + c] = logf(fabsf(f[6 + c]) + FEPS);
    x[14] = sqrtf(f[7]) / (sqrtf(f[6]) + FEPS);
    float sn, cs;
    sincosf(f[5], &sn, &cs);
    x[15] = cs; x[16] = sn;
    x[17] = 0.0f; x[18] = 0.0f; x[19] = 0.0f;

    #pragma unroll
    for (int k = 0; k < 20; ++k) Xs[tid][k] = x[k];
    Ds[tid] = diff[nodeC];

    if (node < N) {
        int p = parent[node];
        par0[node] = (p < 0) ? N : p;
    }
    __syncthreads();

    const float b = bias[0];

    // ---- phase 2: two groups of 16 rows per wave, 5 chained K=4 WMMAs ----
    const int lane    = tid & 31;
    const int waveRow = (tid >> 5) * 32;     // this wave's LDS row base
    const int half    = lane >> 4;           // 0: K=4j+{0,1}, 1: K=4j+{2,3}
    const int m       = lane & 15;           // A-matrix row M

    // B operand (same for both groups): every column holds the weight vector.
    // 32-bit 4x16 layout: lanes 0-15 hold K=4j+{0,1}, lanes 16-31 K=4j+{2,3}.
    v2f bvec[5];
    #pragma unroll
    for (int j = 0; j < 5; ++j)
        bvec[j] = *(const v2f*)&Ws[4 * j + 2 * half];   // ds_load_b64

    #pragma unroll
    for (int g = 0; g < 2; ++g) {
        v8f c = {0.f, 0.f, 0.f, 0.f, 0.f, 0.f, 0.f, 0.f};
        const int row = waveRow + g * 16 + m;
        #pragma unroll
        for (int j = 0; j < 5; ++j) {
            // 32-bit A 16x4 layout, K-pair per lane half; 8B-aligned -> b64 load
            v2f a = *(const v2f*)&Xs[row][4 * j + 2 * half];
            c = __builtin_amdgcn_wmma_f32_16x16x4_f32(
                    /*neg_a=*/false, a, /*neg_b=*/false, bvec[j],
                    /*c_mod=*/(short)0, c, /*reuse_a=*/false, /*reuse_b=*/false);
        }
        // ---- phase 3: extract column N=0 (lane 0 -> M=0..7, lane 16 -> M=8..15)
        if ((lane & 15) == 0) {
            const int mbase = g * 16 + half * 8;
            #pragma unroll
            for (int r = 0; r < 8; ++r) {
                const int rowOut  = waveRow + mbase + r;
                const int nodeOut = blockIdx.x * 256 + rowOut;
                if (nodeOut < N) {
                    float lin = c[r] + b;
                    float sg  = 1.0f / (1.0f + expf(-lin));
                    term0[nodeOut] = Ds[rowOut] * sg;
                }
            }
        }
    }
}

// ---------------------------------------------------------------------------
// Kernel 2: one pointer-jumping step (L2-resident: ~32 MB working set)
// ---------------------------------------------------------------------------
__global__ __launch_bounds__(256) void k_jump(
    const float* __restrict__ termIn,
    const int*   __restrict__ parIn,
    float* __restrict__ termOut,
    int*   __restrict__ parOut,
    int n1)
{
    int i = blockIdx.x * blockDim.x + threadIdx.x;
    if (i < n1) {
        int p = parIn[i];
        termOut[i] = termIn[i] + termIn[p];
        parOut[i]  = parIn[p];
    }
}

// ---------------------------------------------------------------------------
// Kernel 3: final pixel gather, vectorized int4/float4 for stream traffic
// ---------------------------------------------------------------------------
__global__ __launch_bounds__(256) void k_gather(
    const float* __restrict__ vals,
    const int*   __restrict__ pix,
    float* __restrict__ out,
    int n)
{
    int i  = blockIdx.x * blockDim.x + threadIdx.x;
    int i4 = i * 4;
    if (i4 + 3 < n) {
        int4 p = *(const int4*)(pix + i4);
        float4 r;
        r.x = vals[p.x]; r.y = vals[p.y]; r.z = vals[p.z]; r.w = vals[p.w];
        *(float4*)(out + i4) = r;
    } else {
        for (; i4 < n; ++i4) out[i4] = vals[pix[i4]];
    }
}

// ---------------------------------------------------------------------------
extern "C" void kernel_launch(void* const* d_in, const int* in_sizes, int n_in,
                              void* d_out, int out_size, void* d_ws, size_t ws_size,
                              hipStream_t stream)
{
    const float* diff   = (const float*)d_in[0];
    const float* attr   = (const float*)d_in[1];
    const float* weight = (const float*)d_in[2];
    const float* bias   = (const float*)d_in[3];
    const int*   parent = (const int*)d_in[4];
    const int*   pixel  = (const int*)d_in[5];
    float* out = (float*)d_out;

    const int N    = in_sizes[0];
    const int npix = out_size;

    const size_t n1    = (size_t)N + 1;
    const size_t n1pad = (n1 + 255) & ~(size_t)255;   // keep buffers aligned

    char* ws = (char*)d_ws;
    float* tbuf[2] = { (float*)(ws),
                       (float*)(ws + 2 * n1pad * 4) };
    int*   pbuf[2] = { (int*)  (ws + 1 * n1pad * 4),
                       (int*)  (ws + 3 * n1pad * 4) };

    // Phase A: term + par init (WMMA GEMV inside)
    int blocks1 = (N + 255) / 256;
    k_term_par<<<blocks1, 256, 0, stream>>>(diff, attr, weight, bias, parent,
                                            tbuf[0], pbuf[0], N);

    // Phase B: ceil(log2(N)) pointer-jumping sweeps (ping-pong)
    int steps = 0;
    while ((1LL << steps) < (long long)N) ++steps;
    int jb = (int)((n1 + 255) / 256);
    int cur = 0;
    for (int s = 0; s < steps; ++s) {
        k_jump<<<jb, 256, 0, stream>>>(tbuf[cur], pbuf[cur],
                                       tbuf[1 - cur], pbuf[1 - cur], (int)n1);
        cur ^= 1;
    }

    // Phase C: pixel gather
    int gb = ((npix + 3) / 4 + 255) / 256;
    k_gather<<<gb, 256, 0, stream>>>(tbuf[cur], pixel, out, npix);
}